// Embed_and_Prep_55207509623401
// MI455X (gfx1250) — compile-verified
//
#include <hip/hip_runtime.h>
#include <math.h>

// Problem constants (match reference).
#define NPTS 262144
#define NS1  8192
#define NS2  256
#define SPAD 64
#define FEAT 11
#define H1   128
#define H2   256
#define H3   512
#define DIM  384
#define PH   128

typedef __attribute__((ext_vector_type(2))) float v2f;
typedef __attribute__((ext_vector_type(8))) float v8f;

// ---- LDS layout (floats), dynamic shared. Total 29504 floats = 118016 B. ----
// region0: X(32x12)=384, t(32x130)=4160          [dead after f]
// region1: f(32x258)=8256, g(256)                [dead after u]
// region2: u(32x514)=16448                       [dead after y]
// y(32x384)=12288 aliases region0+part of region1 (<=12800) after f is dead.
#define OFF_X 0
#define OFF_T 384
#define OFF_F 4544
#define OFF_G 12800
#define OFF_U 13056
#define OFF_Y 0
#define LDT 130
#define LDF 258
#define LDU 514
#define LDY 384
#define SMEM_FLOATS 29504

#define WMMA_F32(acc, a, b) \
  acc = __builtin_amdgcn_wmma_f32_16x16x4_f32(false, (a), false, (b), (short)0, (acc), false, false)

__global__ __launch_bounds__(256)
void tok_embed_kernel(const float* __restrict__ X,
                      const float* __restrict__ W1, const float* __restrict__ b1,
                      const float* __restrict__ W2, const float* __restrict__ b2,
                      const float* __restrict__ W3, const float* __restrict__ b3,
                      const float* __restrict__ W4, const float* __restrict__ b4,
                      float* __restrict__ out_tok, float* __restrict__ out_pos)
{
  extern __shared__ float sm[];
  const int tid  = threadIdx.x;
  const int lane = tid & 31;
  const int wave = tid >> 5;
  const int half = lane >> 4;     // 0: lanes 0-15, 1: lanes 16-31
  const int l16  = lane & 15;
  const int sp   = blockIdx.x;    // sp1 index [0, 8192)
  const int s2   = sp >> 5;
  const int rank = sp & 31;

  float* sX = sm + OFF_X;
  float* sT = sm + OFF_T;
  float* sF = sm + OFF_F;
  float* sG = sm + OFF_G;
  float* sU = sm + OFF_U;
  float* sY = sm + OFF_Y;

  // ---- Stage in: X rows (32x11, padded to 12) and W1/b1 (staged in dead sU) ----
  for (int i = tid; i < 32 * FEAT; i += 256) {
    int m = i / FEAT, c = i - m * FEAT;
    sX[m * 12 + c] = X[(sp * 32 + m) * FEAT + c];
  }
  for (int i = tid; i < H1 * FEAT; i += 256) sU[i] = W1[i];
  for (int i = tid; i < H1; i += 256)        sU[H1 * FEAT + i] = b1[i];
  __syncthreads();

  // ---- t = X @ W1^T + b1 : [32,128]  (tiny, K=11 -> VALU) ----
  for (int i = tid; i < 32 * H1; i += 256) {
    int m = i >> 7, j = i & 127;
    float acc = sU[H1 * FEAT + j];
    const float* wr = sU + j * FEAT;
    const float* xr = sX + m * 12;
    #pragma unroll
    for (int c = 0; c < FEAT; ++c) acc += xr[c] * wr[c];
    sT[m * LDT + j] = acc;
  }
  __syncthreads();

  // ---- f = t @ W2^T + b2 : [32,256], K=128, fp32 WMMA ----
  // Pair of N-tiles per wave iteration: shared A-frag, 2 independent acc chains.
  for (int p = wave; p < 16; p += 8) {
    int m0 = (p & 1) * 16;
    int n0 = (p >> 1) * 32;
    v8f acc0 = {}, acc1 = {};
    const float* av  = sT + (m0 + l16) * LDT + 2 * half;
    const float* bw0 = W2 + (n0 + l16) * H1 + 2 * half;   // B[k][n] = W2[n][k]
    const float* bw1 = bw0 + 16 * H1;
    #pragma unroll 4
    for (int k = 0; k < H1; k += 4) {
      v2f a  = *(const v2f*)(av + k);
      v2f b0 = *(const v2f*)(bw0 + k);
      v2f b1 = *(const v2f*)(bw1 + k);
      WMMA_F32(acc0, a, b0);
      WMMA_F32(acc1, a, b1);
    }
    float bias0 = b2[n0 + l16];
    float bias1 = b2[n0 + 16 + l16];
    #pragma unroll
    for (int r = 0; r < 8; ++r) {
      int m = m0 + half * 8 + r;
      sF[m * LDF + n0 + l16]      = acc0[r] + bias0;
      sF[m * LDF + n0 + 16 + l16] = acc1[r] + bias1;
    }
  }
  __syncthreads();

  // ---- g = column-max of f over the 32 points of this superpoint ----
  for (int j = tid; j < H2; j += 256) {
    float mx = sF[j];
    #pragma unroll 4
    for (int m = 1; m < 32; ++m) mx = fmaxf(mx, sF[m * LDF + j]);
    sG[j] = mx;
  }
  __syncthreads();

  // ---- u = relu([g|f] @ W3^T + b3) : [32,512], K=512, fp32 WMMA ----
  for (int p = wave; p < 32; p += 8) {
    int m0 = (p & 1) * 16;
    int n0 = (p >> 1) * 32;
    v8f acc0 = {}, acc1 = {};
    const float* ag  = sG + 2 * half;                     // h[m][k]=g[k], k<256 (broadcast)
    const float* af  = sF + (m0 + l16) * LDF + 2 * half;  // h[m][k]=f[m][k-256]
    const float* bw0 = W3 + (n0 + l16) * H3 + 2 * half;
    const float* bw1 = bw0 + 16 * H3;
    #pragma unroll 4
    for (int k = 0; k < 256; k += 4) {
      v2f a  = *(const v2f*)(ag + k);
      v2f b0 = *(const v2f*)(bw0 + k);
      v2f b1 = *(const v2f*)(bw1 + k);
      WMMA_F32(acc0, a, b0);
      WMMA_F32(acc1, a, b1);
    }
    #pragma unroll 4
    for (int k = 256; k < 512; k += 4) {
      v2f a  = *(const v2f*)(af + (k - 256));
      v2f b0 = *(const v2f*)(bw0 + k);
      v2f b1 = *(const v2f*)(bw1 + k);
      WMMA_F32(acc0, a, b0);
      WMMA_F32(acc1, a, b1);
    }
    float bias0 = b3[n0 + l16];
    float bias1 = b3[n0 + 16 + l16];
    #pragma unroll
    for (int r = 0; r < 8; ++r) {
      int m = m0 + half * 8 + r;
      sU[m * LDU + n0 + l16]      = fmaxf(acc0[r] + bias0, 0.0f);
      sU[m * LDU + n0 + 16 + l16] = fmaxf(acc1[r] + bias1, 0.0f);
    }
  }
  __syncthreads();

  // ---- y = u @ W4^T + b4 : [32,384], K=512, fp32 WMMA ----
  // sY aliases the (now dead) X/t/f region.
  for (int p = wave; p < 24; p += 8) {
    int m0 = (p & 1) * 16;
    int n0 = (p >> 1) * 32;
    v8f acc0 = {}, acc1 = {};
    const float* av  = sU + (m0 + l16) * LDU + 2 * half;
    const float* bw0 = W4 + (n0 + l16) * H3 + 2 * half;
    const float* bw1 = bw0 + 16 * H3;
    #pragma unroll 4
    for (int k = 0; k < H3; k += 4) {
      v2f a  = *(const v2f*)(av + k);
      v2f b0 = *(const v2f*)(bw0 + k);
      v2f b1 = *(const v2f*)(bw1 + k);
      WMMA_F32(acc0, a, b0);
      WMMA_F32(acc1, a, b1);
    }
    float bias0 = b4[n0 + l16];
    float bias1 = b4[n0 + 16 + l16];
    #pragma unroll
    for (int r = 0; r < 8; ++r) {
      int m = m0 + half * 8 + r;
      sY[m * LDY + n0 + l16]      = acc0[r] + bias0;
      sY[m * LDY + n0 + 16 + l16] = acc1[r] + bias1;
    }
  }
  __syncthreads();

  // ---- tok = row-max -> scatter to out[s2, rank, :]; zero pad rows 32+rank ----
  size_t row  = (size_t)(s2 * SPAD + rank) * DIM;
  size_t zrow = (size_t)(s2 * SPAD + 32 + rank) * DIM;
  for (int j = tid; j < DIM; j += 256) {
    float mx = sY[j];
    #pragma unroll 4
    for (int m = 1; m < 32; ++m) mx = fmaxf(mx, sY[m * LDY + j]);
    out_tok[row + j]  = mx;
    out_tok[zrow + j] = 0.0f;
    out_pos[zrow + j] = 0.0f;   // pad rows of pos tensor zeroed here too
  }
}

// ---- Pos_Embed: Linear(3->128) -> exact GELU -> Linear(128->384), scattered ----
__global__ __launch_bounds__(128)
void pos_embed_kernel(const float* __restrict__ C,
                      const float* __restrict__ P1, const float* __restrict__ pb1,
                      const float* __restrict__ P2, const float* __restrict__ pb2,
                      float* __restrict__ out_pos)
{
  __shared__ float ph[PH];
  __shared__ float cc[3];
  const int tid = threadIdx.x;
  const int sp  = blockIdx.x;
  if (tid < 3) cc[tid] = C[sp * 3 + tid];
  __syncthreads();
  {
    const float* pr = P1 + tid * 3;
    float x = cc[0] * pr[0] + cc[1] * pr[1] + cc[2] * pr[2] + pb1[tid];
    ph[tid] = 0.5f * x * (1.0f + erff(x * 0.70710678118654752f));  // exact GELU
  }
  __syncthreads();
  const int s2 = sp >> 5, rank = sp & 31;
  size_t row = (size_t)(s2 * SPAD + rank) * DIM;
  #pragma unroll
  for (int jj = 0; jj < 3; ++jj) {
    int j = tid + jj * PH;
    const float* pr = P2 + j * PH;
    float acc = pb2[j];
    #pragma unroll 8
    for (int k = 0; k < PH; ++k) acc += pr[k] * ph[k];
    out_pos[row + j] = acc;
  }
}

extern "C" void kernel_launch(void* const* d_in, const int* in_sizes, int n_in,
                              void* d_out, int out_size, void* d_ws, size_t ws_size,
                              hipStream_t stream) {
  (void)in_sizes; (void)n_in; (void)out_size; (void)d_ws; (void)ws_size;
  const float* X   = (const float*)d_in[0];
  const float* Cd  = (const float*)d_in[1];
  const float* W1  = (const float*)d_in[2];
  const float* b1  = (const float*)d_in[3];
  const float* W2  = (const float*)d_in[4];
  const float* b2  = (const float*)d_in[5];
  const float* W3  = (const float*)d_in[6];
  const float* b3  = (const float*)d_in[7];
  const float* W4  = (const float*)d_in[8];
  const float* b4  = (const float*)d_in[9];
  const float* P1  = (const float*)d_in[10];
  const float* pb1 = (const float*)d_in[11];
  const float* P2  = (const float*)d_in[12];
  const float* pb2 = (const float*)d_in[13];
  // d_in[14], d_in[15] (segment indices) are i/32 by construction; not needed.

  float* out_tok = (float*)d_out;
  float* out_pos = out_tok + (size_t)NS2 * SPAD * DIM;

  const size_t smem = SMEM_FLOATS * sizeof(float);
  tok_embed_kernel<<<NS1, 256, smem, stream>>>(X, W1, b1, W2, b2, W3, b3, W4, b4,
                                               out_tok, out_pos);
  pos_embed_kernel<<<NS1, 128, 0, stream>>>(Cd, P1, pb1, P2, pb2, out_pos);
}